// GraphConvolutionalNetwork_58720792871767
// MI455X (gfx1250) — compile-verified
//
#include <hip/hip_runtime.h>

typedef __attribute__((ext_vector_type(2))) float v2f;
typedef __attribute__((ext_vector_type(8))) float v8f;
typedef __attribute__((ext_vector_type(4))) unsigned int u32x4;
typedef __attribute__((ext_vector_type(4))) int i32x4;
typedef __attribute__((ext_vector_type(8))) int i32x8;

#define NODES 100000
#define EDGES 1600000
#define DIM   128
#define NCLS  40

#ifndef __has_builtin
#define __has_builtin(x) 0
#endif
#if defined(__AMDGCN__) && __has_builtin(__builtin_amdgcn_tensor_load_to_lds) && \
    __has_builtin(__builtin_amdgcn_s_wait_tensorcnt)
#define USE_TDM 1
#else
#define USE_TDM 0
#endif

// ---------------------------------------------------------------------------
// Y[M x NOUT] = act(X[M x K] @ W[K x NOUT] + b[NOUT]) (+ skip)
// Block: 128 threads = 4 waves. Each wave computes a 16-row x 64-col strip:
// 4 v8f accumulators fed by V_WMMA_F32_16X16X4_F32. One A-fragment load
// (2 VGPRs from global) feeds 4 back-to-back WMMAs against B fragments from
// a K x 64 weight slab in LDS (32 KB), staged by the Tensor Data Mover.
//
// Fragment mapping (wave32):
//   A (16x4 f32): r=lane&15 -> row; grp=lane>>4 selects K pair
//   B (4x16 f32): r -> col within sub-tile t
//   C/D (16x16 f32, v8f): acc[t][i] = Y[mbase + i + 8*grp][n0 + 16*t + r]
// ---------------------------------------------------------------------------
__global__ __launch_bounds__(128) void gemm_wmma_f32(
    const float* __restrict__ X, const float* __restrict__ W,
    const float* __restrict__ bias, const float* __restrict__ skip,
    float* __restrict__ Y, int M, int K, int NOUT, int do_relu)
{
  __shared__ float ldsW[DIM * 64];   // K x 64 weight slab (K <= 128), 32 KB

  const int tid = threadIdx.x;
  const int n0  = blockIdx.y * 64;

#if USE_TDM
  // --- Stage W[0:K, n0:n0+64] -> LDS via Tensor DMA (wave 0 issues).
  // TDM zero-fills reads past tensor_dim0, which implements the NOUT=40
  // column padding for free.
  if (tid < 32) {
    const unsigned lds_off  = (unsigned)(uintptr_t)&ldsW[0];      // low 32b = LDS offset
    const unsigned long long gaddr =
        (unsigned long long)(uintptr_t)(W + n0);                  // tile start (bytes)
    const unsigned td0 = (unsigned)(NOUT - n0);                   // cols remaining from tile start
    const unsigned td1 = (unsigned)K;
    const unsigned long long s0 = (unsigned long long)NOUT;       // row stride (elements)

    u32x4 g0;
    g0.x = 1u;                                                    // count=1, user mode
    g0.y = lds_off;                                               // lds_addr
    g0.z = (unsigned)(gaddr & 0xFFFFFFFFull);                     // global_addr[31:0]
    g0.w = (unsigned)((gaddr >> 32) & 0x01FFFFFFull) | (2u << 30);// addr[56:32] | type=2

    i32x8 g1;
    g1[0] = (int)(2u << 16);                                      // data_size=2 (4B), mask=0
    g1[1] = (int)((td0 & 0xFFFFu) << 16);                         // tensor_dim0[15:0]
    g1[2] = (int)((td0 >> 16) | ((td1 & 0xFFFFu) << 16));         // dim0[31:16] | dim1[15:0]
    g1[3] = (int)((td1 >> 16) | (64u << 16));                     // dim1[31:16] | tile_dim0=64
    g1[4] = (int)((unsigned)K);                                   // tile_dim1=K, tile_dim2=0
    g1[5] = (int)(unsigned)(s0 & 0xFFFFFFFFull);                  // dim0_stride[31:0]
    g1[6] = (int)(unsigned)(s0 >> 32);                            // dim0_stride[47:32] | dim1_stride lo
    g1[7] = 0;
    i32x4 gz4 = {0, 0, 0, 0};
#if __clang_major__ >= 23
    i32x8 gz8 = {0, 0, 0, 0, 0, 0, 0, 0};
    __builtin_amdgcn_tensor_load_to_lds(g0, g1, gz4, gz4, gz8, 0);
#else
    __builtin_amdgcn_tensor_load_to_lds(g0, g1, gz4, gz4, 0);
#endif
    __builtin_amdgcn_s_wait_tensorcnt(0);
  }
  __syncthreads();
#else
  // Fallback: cooperative staged load with zero padding.
  for (int idx = tid; idx < K * 64; idx += 128) {
    const int k = idx >> 6;
    const int c = idx & 63;
    ldsW[idx] = (n0 + c < NOUT) ? W[k * NOUT + n0 + c] : 0.0f;
  }
  __syncthreads();
#endif

  const int wave  = tid >> 5;
  const int lane  = tid & 31;
  const int grp   = lane >> 4;     // 0: lanes 0-15, 1: lanes 16-31
  const int r     = lane & 15;
  const int mbase = blockIdx.x * 64 + wave * 16;
  if (mbase >= M) return;          // wave-uniform: EXEC all-1s for WMMA (M%16==0)

  const float* __restrict__ Xrow = X + (size_t)(mbase + r) * K;

  v8f acc0 = {}, acc1 = {}, acc2 = {}, acc3 = {};
  #pragma unroll 4
  for (int k0 = 0; k0 < K; k0 += 4) {
    const int ka = k0 + 2 * grp;
    v2f a;
    a.x = Xrow[ka];
    a.y = Xrow[ka + 1];
    const float* __restrict__ wrow0 = &ldsW[ka * 64 + r];
    const float* __restrict__ wrow1 = &ldsW[(ka + 1) * 64 + r];
    v2f b0, b1, b2, b3;
    b0.x = wrow0[ 0]; b0.y = wrow1[ 0];
    b1.x = wrow0[16]; b1.y = wrow1[16];
    b2.x = wrow0[32]; b2.y = wrow1[32];
    b3.x = wrow0[48]; b3.y = wrow1[48];
    acc0 = __builtin_amdgcn_wmma_f32_16x16x4_f32(false, a, false, b0, (short)0, acc0, false, false);
    acc1 = __builtin_amdgcn_wmma_f32_16x16x4_f32(false, a, false, b1, (short)0, acc1, false, false);
    acc2 = __builtin_amdgcn_wmma_f32_16x16x4_f32(false, a, false, b2, (short)0, acc2, false, false);
    acc3 = __builtin_amdgcn_wmma_f32_16x16x4_f32(false, a, false, b3, (short)0, acc3, false, false);
  }

  v8f accs[4] = {acc0, acc1, acc2, acc3};
  #pragma unroll
  for (int t = 0; t < 4; ++t) {
    const int col = n0 + t * 16 + r;
    if (col < NOUT) {
      const float bv = bias[col];
      #pragma unroll
      for (int i = 0; i < 8; ++i) {
        const int row = mbase + i + 8 * grp;
        float v = accs[t][i] + bv;
        if (do_relu) v = fmaxf(v, 0.0f);
        if (skip)    v += skip[(size_t)row * NOUT + col];  // h = relu(conv@W+b) + conv
        Y[(size_t)row * NOUT + col] = v;
      }
    }
  }
}

// ---------------------------------------------------------------------------
// conv[receivers[e]] += edge_attr[e] * h[senders[e]]   (fused gather+scatter)
// One wave per EPW-edge chunk; each lane handles 4 contiguous features.
// h (51.2 MB) and conv (51.2 MB) both fit in the 192 MB L2, so gathers and
// the f32 atomics are L2-resident. Software-pipeline the gather: prefetch the
// next sender's row (global_prefetch_b8) while processing the current edge.
// ---------------------------------------------------------------------------
#define EPW 8
__global__ __launch_bounds__(256) void edge_scatter(
    const int* __restrict__ edge_index,   // [2, E]: row 0 = receivers, row 1 = senders
    const float* __restrict__ edge_attr,  // [E]
    const float* __restrict__ h,          // [N, 128]
    float* __restrict__ conv,             // [N, 128]
    int E)
{
  const long long wv = ((long long)blockIdx.x * blockDim.x + threadIdx.x) >> 5;
  const int lane     = threadIdx.x & 31;
  const long long base = wv * EPW;
  if (base >= E) return;
  const long long end = (base + EPW < E) ? base + EPW : E;

  {
    const int s0 = edge_index[(long long)E + base];
    __builtin_prefetch(h + (size_t)s0 * DIM + lane * 4, 0, 1);
  }
  for (long long e = base; e < end; ++e) {
    const int recv = edge_index[e];
    const int send = edge_index[(long long)E + e];
    const float w  = edge_attr[e];
    if (e + 1 < end) {
      const int ns = edge_index[(long long)E + e + 1];
      __builtin_prefetch(h + (size_t)ns * DIM + lane * 4, 0, 1);
    }
    const float4 v = ((const float4*)(h + (size_t)send * DIM))[lane];
    float* dst = conv + (size_t)recv * DIM + lane * 4;
    atomicAdd(dst + 0, w * v.x);
    atomicAdd(dst + 1, w * v.y);
    atomicAdd(dst + 2, w * v.z);
    atomicAdd(dst + 3, w * v.w);
  }
}

__global__ __launch_bounds__(256) void zero_f4(float4* __restrict__ p, long long n4)
{
  const long long i = (long long)blockIdx.x * blockDim.x + threadIdx.x;
  if (i < n4) p[i] = float4{0.0f, 0.0f, 0.0f, 0.0f};
}

// ---------------------------------------------------------------------------
extern "C" void kernel_launch(void* const* d_in, const int* in_sizes, int n_in,
                              void* d_out, int out_size, void* d_ws, size_t ws_size,
                              hipStream_t stream) {
  (void)in_sizes; (void)n_in; (void)out_size; (void)ws_size;

  const float* x      = (const float*)d_in[0];
  const int*   ei     = (const int*)  d_in[1];   // [2, E] (values < 100000)
  const float* ea     = (const float*)d_in[2];   // [E, 1]
  const float* enc_w0 = (const float*)d_in[3];
  const float* enc_b0 = (const float*)d_in[4];
  const float* enc_w1 = (const float*)d_in[5];
  const float* enc_b1 = (const float*)d_in[6];
  const float* hop_w  = (const float*)d_in[7];   // [3, 128, 128]
  const float* hop_b  = (const float*)d_in[8];   // [3, 128]
  const float* dec_w0 = (const float*)d_in[9];
  const float* dec_b0 = (const float*)d_in[10];
  const float* dec_w1 = (const float*)d_in[11];  // [128, 40]
  const float* dec_b1 = (const float*)d_in[12];
  float* out = (float*)d_out;

  const size_t sz = (size_t)NODES * DIM;
  float* bufA = (float*)d_ws;          // ping
  float* bufB = bufA + sz;             // pong
  float* bufC = bufB + sz;             // conv accumulator

  const dim3 blk(128);
  const dim3 g128((NODES + 63) / 64, DIM / 64);        // (1563, 2)
  const dim3 g40 ((NODES + 63) / 64, (NCLS + 63) / 64); // (1563, 1)

  const long long n4 = (long long)sz / 4;
  const int zgrid = (int)((n4 + 255) / 256);
  const long long ewaves = ((long long)EDGES + EPW - 1) / EPW;
  const int egrid = (int)((ewaves * 32 + 255) / 256);

  // Encoder: h = relu(relu(x@W0+b0)@W1+b1)
  gemm_wmma_f32<<<g128, blk, 0, stream>>>(x,    enc_w0, enc_b0, nullptr, bufA, NODES, DIM, DIM, 1);
  gemm_wmma_f32<<<g128, blk, 0, stream>>>(bufA, enc_w1, enc_b1, nullptr, bufB, NODES, DIM, DIM, 1);

  // 3 message-passing hops: conv = scatter_add(edge_attr * h[send]);
  // h = relu(conv@Wi+bi) + conv
  float* h     = bufB;
  float* other = bufA;
  for (int i = 0; i < 3; ++i) {
    zero_f4<<<zgrid, 256, 0, stream>>>((float4*)bufC, n4);
    edge_scatter<<<egrid, 256, 0, stream>>>(ei, ea, h, bufC, EDGES);
    gemm_wmma_f32<<<g128, blk, 0, stream>>>(bufC, hop_w + (size_t)i * DIM * DIM,
                                            hop_b + (size_t)i * DIM,
                                            /*skip=*/bufC, other, NODES, DIM, DIM, 1);
    float* t2 = h; h = other; other = t2;
  }

  // Decoder: out = relu(h@W0+b0)@W1+b1
  gemm_wmma_f32<<<g128, blk, 0, stream>>>(h,     dec_w0, dec_b0, nullptr, other, NODES, DIM, DIM, 1);
  gemm_wmma_f32<<<g40,  blk, 0, stream>>>(other, dec_w1, dec_b1, nullptr, out,   NODES, DIM, NCLS, 0);
}